// HeteroGAT_57793079935346
// MI455X (gfx1250) — compile-verified
//
#include <hip/hip_runtime.h>
#include <math.h>

typedef float v2f __attribute__((ext_vector_type(2)));
typedef float v8f __attribute__((ext_vector_type(8)));

// ---------------------------------------------------------------------------
// float atomic max via sign-magnitude ordering trick (init value must be -inf)
// ---------------------------------------------------------------------------
__device__ inline void atomicMaxF(float* addr, float val) {
  if (val >= 0.f) {
    atomicMax((int*)addr, __float_as_int(val));
  } else {
    atomicMin((unsigned int*)addr, __float_as_uint(val));
  }
}

__global__ void fill_f32_kernel(float* __restrict__ p, float v, int n) {
  int t = blockIdx.x * blockDim.x + threadIdx.x;
  if (t < n) p[t] = v;
}

// Wa[k,h] = sum_c W[k, h*C+c] * a[h,c]   (folds attention vector through W)
__global__ void make_wa_kernel(const float* __restrict__ W, const float* __restrict__ a,
                               float* __restrict__ Wa, int K, int H, int C) {
  int t = blockIdx.x * blockDim.x + threadIdx.x;
  if (t >= K * H) return;
  int k = t / H, h = t % H;
  float s = 0.f;
  for (int c = 0; c < C; ++c) s += W[(long)k * (H * C) + h * C + c] * a[h * C + c];
  Wa[t] = s;
}

// al[n,h] = sum_k X[n,k] * Wa[k,h]
__global__ void att_logit_kernel(const float* __restrict__ X, const float* __restrict__ Wa,
                                 float* __restrict__ out, int Nn, int K, int H) {
  int t = blockIdx.x * blockDim.x + threadIdx.x;
  if (t >= Nn * H) return;
  int n = t / H, h = t % H;
  const float* xr = X + (long)n * K;
  float s = 0.f;
  for (int k = 0; k < K; ++k) s += xr[k] * Wa[k * H + h];
  out[t] = s;
}

// ---------------------------------------------------------------------------
// f32 WMMA GEMM: C[M,N] = A[M,K] @ B[K,N], row-major.
// One wave per (16 rows) x (32 cols): two 16x16 WMMA tiles sharing the A
// fragment. K *and* N are compile-time -> after full unroll every load is
// base + constant IOFFSET (no per-iteration address math). OOB A rows are
// handled by clamping the row index (they only feed OOB D rows, never stored),
// so EXEC stays all-ones around the WMMAs.
// ---------------------------------------------------------------------------
template <int K, int N>
__global__ void gemm_wmma_kernel(const float* __restrict__ A, const float* __restrict__ B,
                                 float* __restrict__ C, int M) {
  const int tn = blockIdx.x * 2;     // two 16-wide N tiles
  const int tm = blockIdx.y;
  const int lane = threadIdx.x;      // blockDim.x == 32 (one wave)
  const int half = lane >> 4;        // 0 or 1
  const int l16  = lane & 15;
  int arow = tm * 16 + l16;
  arow = arow < M ? arow : (M - 1);  // clamp; OOB rows never stored
  // lane-fixed base pointers; all loop offsets become immediates
  const float* __restrict__ Ab = A + (size_t)arow * K + 2 * half;
  const float* __restrict__ Bb = B + (size_t)(2 * half) * N + tn * 16 + l16;
  v8f acc0 = {}, acc1 = {};
#pragma unroll
  for (int k0 = 0; k0 < K; k0 += 4) {
    v2f a = *(const v2f*)(Ab + k0);            // global_load_b64, imm offset
    v2f b0, b1;
    b0.x = Bb[k0 * N];
    b0.y = Bb[k0 * N + N];
    b1.x = Bb[k0 * N + 16];
    b1.y = Bb[k0 * N + N + 16];
    acc0 = __builtin_amdgcn_wmma_f32_16x16x4_f32(false, a, false, b0, (short)0, acc0,
                                                 false, false);
    acc1 = __builtin_amdgcn_wmma_f32_16x16x4_f32(false, a, false, b1, (short)0, acc1,
                                                 false, false);
  }
  const int bcol0 = tn * 16 + l16;
#pragma unroll
  for (int v = 0; v < 8; ++v) {
    const int row = tm * 16 + v + 8 * half;
    if (row < M) {
      C[(size_t)row * N + bcol0]      = acc0[v];
      C[(size_t)row * N + bcol0 + 16] = acc1[v];
    }
  }
}

// ---------------------------------------------------------------------------
// Edge passes
// ---------------------------------------------------------------------------
__global__ void edge_logits_kernel(const int* __restrict__ src, const int* __restrict__ dst,
                                   const float* __restrict__ als, const float* __restrict__ ald,
                                   float* __restrict__ e, float* __restrict__ m,
                                   int E, int H) {
  int t = blockIdx.x * blockDim.x + threadIdx.x;
  if (t >= E * H) return;
  int i = t / H, h = t % H;
  float v = als[src[i] * H + h] + ald[dst[i] * H + h];
  v = v > 0.f ? v : 0.2f * v;                 // leaky_relu(0.2)
  e[t] = v;
  atomicMaxF(&m[dst[i] * H + h], v);
}

__global__ void edge_exp_kernel(const int* __restrict__ dst, const float* __restrict__ m,
                                float* __restrict__ e, float* __restrict__ den,
                                int E, int H) {
  int t = blockIdx.x * blockDim.x + threadIdx.x;
  if (t >= E * H) return;
  int i = t / H, h = t % H;
  float ex = expf(e[t] - m[dst[i] * H + h]);
  e[t] = ex;
  atomicAdd(&den[dst[i] * H + h], ex);
}

// One block per edge; blockDim == H*C (256 for layer 1, 64 for layer 2).
__global__ void edge_scatter_kernel(const int* __restrict__ src, const int* __restrict__ dst,
                                    const float* __restrict__ e, const float* __restrict__ den,
                                    const float* __restrict__ hsrc, float* __restrict__ acc,
                                    int H, int C) {
  const int i = blockIdx.x;
  const int tid = threadIdx.x;                // == h*C + c
  const int h = tid / C;
  const int s = src[i], d = dst[i];
  const float alpha = e[(long)i * H + h] / (den[(long)d * H + h] + 1e-16f);
  const float msg = alpha * hsrc[(long)s * (H * C) + tid];
  atomicAdd(&acc[(long)d * (H * C) + tid], msg);
}

// layer-1 finalize: mean over 2 relations + bias, ELU in place
__global__ void finalize1_kernel(float* __restrict__ acc, const float* __restrict__ ba,
                                 const float* __restrict__ bb, int Nn, int D) {
  long t = (long)blockIdx.x * blockDim.x + threadIdx.x;
  if (t >= (long)Nn * D) return;
  int j = (int)(t % D);
  float v = 0.5f * (acc[t] + ba[j] + bb[j]);
  acc[t] = v > 0.f ? v : (expf(v) - 1.f);     // ELU
}

// layer-2 finalize: mean + bias, then L2 row-normalize. One wave per node, D=64.
__global__ void finalize2_kernel(const float* __restrict__ acc, const float* __restrict__ ba,
                                 const float* __restrict__ bb, float* __restrict__ out,
                                 int Nn) {
  int n = blockIdx.x;
  int lane = threadIdx.x;                     // 32 lanes
  float v0 = 0.5f * (acc[(long)n * 64 + lane]      + ba[lane]      + bb[lane]);
  float v1 = 0.5f * (acc[(long)n * 64 + 32 + lane] + ba[32 + lane] + bb[32 + lane]);
  float ss = v0 * v0 + v1 * v1;
#pragma unroll
  for (int o = 16; o > 0; o >>= 1) ss += __shfl_xor(ss, o, 32);
  float norm = sqrtf(ss);
  norm = norm > 1e-12f ? norm : 1e-12f;
  out[(long)n * 64 + lane]      = v0 / norm;
  out[(long)n * 64 + 32 + lane] = v1 / norm;
}

// ---------------------------------------------------------------------------
// Host orchestration
// ---------------------------------------------------------------------------
extern "C" void kernel_launch(void* const* d_in, const int* in_sizes, int n_in,
                              void* d_out, int out_size, void* d_ws, size_t ws_size,
                              hipStream_t stream) {
  (void)in_sizes; (void)n_in; (void)out_size; (void)ws_size;

  static const int NT[3]   = {40000, 15000, 8000};
  static const int RS[6]   = {0, 1, 0, 1, 2, 2};   // src type per relation
  static const int RD[6]   = {1, 2, 2, 0, 1, 0};   // dst type per relation
  static const int EN[6]   = {400000, 300000, 300000, 400000, 300000, 300000};
  // relations feeding each dst type
  static const int RELA[3] = {3, 0, 1};
  static const int RELB[3] = {5, 4, 2};

  const float* x[3] = {(const float*)d_in[0], (const float*)d_in[1], (const float*)d_in[2]};
  const int* eptr[6];
  for (int r = 0; r < 6; ++r) eptr[r] = (const int*)d_in[3 + r];
  const float* W1  = (const float*)d_in[9];
  const float* as1 = (const float*)d_in[10];
  const float* ad1 = (const float*)d_in[11];
  const float* b1  = (const float*)d_in[12];
  const float* W2  = (const float*)d_in[13];
  const float* as2 = (const float*)d_in[14];
  const float* ad2 = (const float*)d_in[15];
  const float* b2  = (const float*)d_in[16];
  float* out = (float*)d_out;

  // ---- workspace bump allocator (floats) ----
  float* ws = (float*)d_ws;
  size_t off = 0;
  float* acc1[3];
  for (int t = 0; t < 3; ++t) { acc1[t] = ws + off; off += (size_t)NT[t] * 256; }
  float* acc2[3];
  for (int t = 0; t < 3; ++t) { acc2[t] = ws + off; off += (size_t)NT[t] * 64; }
  float* hsrc = ws + off; off += (size_t)40000 * 256;   // max Ns * max D, reused per relation
  float* als  = ws + off; off += (size_t)40000 * 4;
  float* ald  = ws + off; off += (size_t)40000 * 4;
  float* mbuf = ws + off; off += (size_t)40000 * 4;
  float* den  = ws + off; off += (size_t)40000 * 4;
  float* ebuf = ws + off; off += (size_t)400000 * 4;    // max E * max H
  float* waS  = ws + off; off += 256 * 4;
  float* waD  = ws + off; off += 256 * 4;

  // zero the relation-sum accumulators
  for (int t = 0; t < 3; ++t) {
    hipMemsetAsync(acc1[t], 0, (size_t)NT[t] * 256 * sizeof(float), stream);
    hipMemsetAsync(acc2[t], 0, (size_t)NT[t] * 64 * sizeof(float), stream);
  }

  const float NEG_INF = -__builtin_huge_valf();

  // ======================= Layer 1: 128 -> 4x64 (concat) ====================
  for (int r = 0; r < 6; ++r) {
    const int s = RS[r], d = RD[r];
    const int Ns = NT[s], Nd = NT[d], E = EN[r];
    const int K = 128, H = 4, C = 64, D = H * C;  // D = 256
    const float* Wr = W1 + (size_t)r * K * D;
    const int* srcI = eptr[r];
    const int* dstI = eptr[r] + E;

    make_wa_kernel<<<(K * H + 63) / 64, 64, 0, stream>>>(Wr, as1 + (size_t)r * H * C, waS, K, H, C);
    make_wa_kernel<<<(K * H + 63) / 64, 64, 0, stream>>>(Wr, ad1 + (size_t)r * H * C, waD, K, H, C);

    gemm_wmma_kernel<128, 256><<<dim3(256 / 32, (Ns + 15) / 16), 32, 0, stream>>>(x[s], Wr, hsrc, Ns);

    att_logit_kernel<<<(Ns * H + 255) / 256, 256, 0, stream>>>(x[s], waS, als, Ns, K, H);
    att_logit_kernel<<<(Nd * H + 255) / 256, 256, 0, stream>>>(x[d], waD, ald, Nd, K, H);

    fill_f32_kernel<<<(Nd * H + 255) / 256, 256, 0, stream>>>(mbuf, NEG_INF, Nd * H);
    hipMemsetAsync(den, 0, (size_t)Nd * H * sizeof(float), stream);

    edge_logits_kernel<<<(E * H + 255) / 256, 256, 0, stream>>>(srcI, dstI, als, ald, ebuf, mbuf, E, H);
    edge_exp_kernel<<<(E * H + 255) / 256, 256, 0, stream>>>(dstI, mbuf, ebuf, den, E, H);
    edge_scatter_kernel<<<E, H * C, 0, stream>>>(srcI, dstI, ebuf, den, hsrc, acc1[d], H, C);
  }
  for (int t = 0; t < 3; ++t) {
    long n = (long)NT[t] * 256;
    finalize1_kernel<<<(unsigned)((n + 255) / 256), 256, 0, stream>>>(
        acc1[t], b1 + (size_t)RELA[t] * 256, b1 + (size_t)RELB[t] * 256, NT[t], 256);
  }

  // ======================= Layer 2: 256 -> 64 (mean heads) ==================
  for (int r = 0; r < 6; ++r) {
    const int s = RS[r], d = RD[r];
    const int Ns = NT[s], Nd = NT[d], E = EN[r];
    const int K = 256, H = 1, C = 64;
    const float* Wr = W2 + (size_t)r * K * 64;
    const int* srcI = eptr[r];
    const int* dstI = eptr[r] + E;

    make_wa_kernel<<<(K * H + 63) / 64, 64, 0, stream>>>(Wr, as2 + (size_t)r * C, waS, K, H, C);
    make_wa_kernel<<<(K * H + 63) / 64, 64, 0, stream>>>(Wr, ad2 + (size_t)r * C, waD, K, H, C);

    gemm_wmma_kernel<256, 64><<<dim3(64 / 32, (Ns + 15) / 16), 32, 0, stream>>>(acc1[s], Wr, hsrc, Ns);

    att_logit_kernel<<<(Ns * H + 255) / 256, 256, 0, stream>>>(acc1[s], waS, als, Ns, K, H);
    att_logit_kernel<<<(Nd * H + 255) / 256, 256, 0, stream>>>(acc1[d], waD, ald, Nd, K, H);

    fill_f32_kernel<<<(Nd * H + 255) / 256, 256, 0, stream>>>(mbuf, NEG_INF, Nd * H);
    hipMemsetAsync(den, 0, (size_t)Nd * H * sizeof(float), stream);

    edge_logits_kernel<<<(E * H + 255) / 256, 256, 0, stream>>>(srcI, dstI, als, ald, ebuf, mbuf, E, H);
    edge_exp_kernel<<<(E * H + 255) / 256, 256, 0, stream>>>(dstI, mbuf, ebuf, den, E, H);
    edge_scatter_kernel<<<E, H * C, 0, stream>>>(srcI, dstI, ebuf, den, hsrc, acc2[d], H, C);
  }

  // finalize + L2 normalize into d_out (circ | mi | dis, each N x 64)
  float* outp = out;
  for (int t = 0; t < 3; ++t) {
    finalize2_kernel<<<NT[t], 32, 0, stream>>>(
        acc2[t], b2 + (size_t)RELA[t] * 64, b2 + (size_t)RELB[t] * 64, outp, NT[t]);
    outp += (size_t)NT[t] * 64;
  }
}